// SSKernel_55765855371366
// MI455X (gfx1250) — compile-verified
//
#include <hip/hip_runtime.h>
#include <hip/hip_bf16.h>
#include <math.h>

// ---------------------------------------------------------------------------
// S4 SSKernelNPLR forward for MI455X (gfx1250).
//   Stage A: fill TRANSPOSED DFT basis Bt (2048 x 2080) into workspace.
//   Stage B: Cauchy + rank-1 Woodbury -> padded A matrix (512 x 2080)
//            A[:, 0:1025] = Re(k_f),  A[:, 1040:2065] = Im(k_f), pads = 0
//   Stage C: irfft as fp32 WMMA GEMM: out(512x2048) = A(512xKP) @ Bt^T
//            (one v_wmma_f32_16x16x4_f32 per 4 K-steps; b64 loads for both
//             operands thanks to the transposed basis; global_prefetch ahead)
// ---------------------------------------------------------------------------

#define HH    512
#define NN2   32
#define LL    2048
#define MM    1025          // L/2 + 1
#define MPAD  1040          // M padded to multiple of 16
#define KP    2080          // 2*MPAD  (GEMM K dimension)

typedef float v2f __attribute__((ext_vector_type(2)));
typedef float v8f __attribute__((ext_vector_type(8)));

struct cpx { float re, im; };
__device__ __forceinline__ cpx cmul(cpx a, cpx b) { return {a.re*b.re - a.im*b.im, a.re*b.im + a.im*b.re}; }
__device__ __forceinline__ cpx cadd(cpx a, cpx b) { return {a.re + b.re, a.im + b.im}; }
__device__ __forceinline__ cpx csub(cpx a, cpx b) { return {a.re - b.re, a.im - b.im}; }
__device__ __forceinline__ cpx crcp(cpx a) {
    float inv = 1.0f / (a.re*a.re + a.im*a.im);
    return {a.re*inv, -a.im*inv};
}
__device__ __forceinline__ cpx cdiv(cpx a, cpx b) { return cmul(a, crcp(b)); }
__device__ __forceinline__ cpx cconj(cpx a) { return {a.re, -a.im}; }

// ---------------------------------------------------------------------------
// Stage A: transposed basis Bt[l][r], l in [0,2048), r in [0,KP).
//  r in [0,1025)        : ( s_m / L) * cos(2*pi*r*l/L)
//  r in [1025,1040)     : 0 (pad)
//  r in [1040,2065)     : (-s_m / L) * sin(2*pi*(r-1040)*l/L)
//  r in [2065,2080)     : 0 (pad)
//  s_m = 1 for m==0 or m==1024 (Nyquist), else 2.
// One block per l; threads sweep r.
// ---------------------------------------------------------------------------
__global__ void s4_fill_basis_t(float* __restrict__ Bt) {
    const int l = blockIdx.x;
    const float step = 6.283185307179586f / (float)LL;
    float* row = Bt + (size_t)l * KP;
    for (int r = threadIdx.x; r < KP; r += blockDim.x) {
        float val = 0.0f;
        if (r < MM) {
            int m = r;
            int t = (m * l) & (LL - 1);              // phase periodic in L
            float sc = ((m == 0) || (m == LL / 2)) ? 1.0f : 2.0f;
            val = (sc * (1.0f / (float)LL)) * __cosf((float)t * step);
        } else if (r >= MPAD && r < MPAD + MM) {
            int m = r - MPAD;
            int t = (m * l) & (LL - 1);
            float sc = ((m == 0) || (m == LL / 2)) ? 1.0f : 2.0f;
            val = -(sc * (1.0f / (float)LL)) * __sinf((float)t * step);
        }
        row[r] = val;
    }
}

// ---------------------------------------------------------------------------
// Stage B: one block per h.  Threads 0..31 build per-n data into LDS, then
// every thread evaluates the Cauchy sum + Woodbury for a strided set of m.
// ---------------------------------------------------------------------------
__global__ void s4_cauchy_woodbury(const float* __restrict__ log_dt,
                                   const float* __restrict__ log_w_real,
                                   const float* __restrict__ w_imag,
                                   const float* __restrict__ Bin,
                                   const float* __restrict__ Cin,
                                   const float* __restrict__ Pin,
                                   float* __restrict__ Amat) {
    __shared__ float2 sh_w[NN2];
    __shared__ float2 sh_v00[NN2], sh_v01[NN2], sh_v10[NN2], sh_v11[NN2];

    const int h = blockIdx.x;
    const float dt = __expf(log_dt[h]);

    if (threadIdx.x < NN2) {
        int n = threadIdx.x;
        int base = (h * NN2 + n) * 2;
        cpx Bc = {Bin[base], Bin[base + 1]};
        cpx Cc = {Cin[base], Cin[base + 1]};   // channels == 1
        cpx Pc = {Pin[base], Pin[base + 1]};   // rank == 1
        cpx Qc = cconj(Pc);
        float wre = -__expf(log_w_real[h * NN2 + n]) * dt;
        float wim = w_imag[h * NN2 + n] * dt;
        sh_w[n] = make_float2(wre, wim);
        cpx v00 = cmul(Bc, Cc);  sh_v00[n] = make_float2(v00.re, v00.im);
        cpx v01 = cmul(Bc, Qc);  sh_v01[n] = make_float2(v01.re, v01.im);
        cpx v10 = cmul(Pc, Cc);  sh_v10[n] = make_float2(v10.re, v10.im);
        cpx v11 = cmul(Pc, Qc);  sh_v11[n] = make_float2(v11.re, v11.im);
    }
    __syncthreads();

    const float step = 6.283185307179586f / (float)LL;
    for (int m = threadIdx.x; m < MPAD; m += blockDim.x) {
        float* rowRe = Amat + (size_t)h * KP + m;
        float* rowIm = rowRe + MPAD;
        if (m >= MM) {            // zero pad columns (ws is poisoned!)
            *rowRe = 0.0f; *rowIm = 0.0f;
            continue;
        }
        float s, c;
        __sincosf((float)m * step, &s, &c);
        cpx om = {c, -s};                         // exp(-2*pi*i*m/L)
        cpx d1 = {1.0f + om.re, om.im};           // 1 + omega
        cpx z  = cdiv({2.0f * (1.0f - om.re), -2.0f * om.im}, d1);

        cpx a00 = {0, 0}, a01 = {0, 0}, a10 = {0, 0}, a11 = {0, 0};
        for (int n = 0; n < NN2; ++n) {
            float2 wf = sh_w[n];
            cpx den1 = crcp({z.re - wf.x, z.im - wf.y});   // 1/(z - w)
            cpx den2 = crcp({z.re - wf.x, z.im + wf.y});   // 1/(z - conj(w))
            float2 t;
            t = sh_v00[n]; { cpx v = {t.x, t.y};
                a00 = cadd(a00, cadd(cmul(v, den1), cmul(cconj(v), den2))); }
            t = sh_v01[n]; { cpx v = {t.x, t.y};
                a01 = cadd(a01, cadd(cmul(v, den1), cmul(cconj(v), den2))); }
            t = sh_v10[n]; { cpx v = {t.x, t.y};
                a10 = cadd(a10, cadd(cmul(v, den1), cmul(cconj(v), den2))); }
            t = sh_v11[n]; { cpx v = {t.x, t.y};
                a11 = cadd(a11, cadd(cmul(v, den1), cmul(cconj(v), den2))); }
        }
        // r = r * dt
        a00.re *= dt; a00.im *= dt;  a01.re *= dt; a01.im *= dt;
        a10.re *= dt; a10.im *= dt;  a11.re *= dt; a11.im *= dt;

        // rank-1 Woodbury:  kf = r00 - r01*r10 / (1 + r11)
        cpx kf = csub(a00, cdiv(cmul(a01, a10), {1.0f + a11.re, a11.im}));
        // kf *= 2/(1+omega)
        kf = cmul(kf, cdiv({2.0f, 0.0f}, d1));

        *rowRe = kf.re;
        *rowIm = kf.im;
    }
}

// ---------------------------------------------------------------------------
// Stage C: fp32 WMMA GEMM  out(512 x 2048) = A(512 x KP) @ B(KP x 2048),
// with B supplied transposed (Bt: 2048 x KP) so each B fragment is one
// aligned b64 load.
// f32 16x16x4 A layout (ISA 7.12.2): lanes 0-15 hold K0(V0),K1(V1);
// lanes 16-31 hold K2(V0),K3(V1).  B layout mirrored.  C/D: VGPR i ->
// row M=i (lanes 0-15) / M=i+8 (lanes 16-31), N = lane&15.
// ---------------------------------------------------------------------------
__global__ void s4_irfft_wmma(const float* __restrict__ Amat,
                              const float* __restrict__ Bt,
                              float* __restrict__ out) {
    const int wave = (blockIdx.x * blockDim.x + threadIdx.x) >> 5; // 0..4095
    const int lane = threadIdx.x & 31;
    const int tM = wave >> 7;      // 0..31  (512/16 row tiles)
    const int tN = wave & 127;     // 0..127 (2048/16 col tiles)
    const int mBase = tM * 16;
    const int nBase = tN * 16;

    const int khalf = (lane >> 4) * 2;        // 0 for lanes 0-15, 2 for 16-31
    const int mrow  = mBase + (lane & 15);
    const int ncol  = nBase + (lane & 15);

    const float* arow = Amat + (size_t)mrow * KP + khalf;  // A[m][k+khalf ...]
    const float* brow = Bt   + (size_t)ncol * KP + khalf;  // B[k+khalf][n] ...

    v8f acc = {};
    for (int kk = 0; kk < KP; kk += 32) {     // KP = 65 * 32
        // pull the streams ~1KB ahead (lowers to global_prefetch_b8;
        // speculative past-the-end prefetch is silently dropped)
        __builtin_prefetch(arow + kk + 256, 0, 1);
        __builtin_prefetch(brow + kk + 256, 0, 1);
#pragma unroll
        for (int k = kk; k < kk + 32; k += 4) {
            v2f a = *(const v2f*)(arow + k);   // A[m][k+khalf], A[m][k+khalf+1]
            v2f b = *(const v2f*)(brow + k);   // B[k+khalf][n], B[k+khalf+1][n]
            acc = __builtin_amdgcn_wmma_f32_16x16x4_f32(
                      /*neg_a=*/false, a, /*neg_b=*/false, b,
                      /*c_mod=*/(short)0, acc,
                      /*reuse_a=*/false, /*reuse_b=*/false);
        }
    }

    const int orow = mBase + ((lane < 16) ? 0 : 8);
    const int ocol = nBase + (lane & 15);
#pragma unroll
    for (int i = 0; i < 8; ++i)
        out[(size_t)(orow + i) * LL + ocol] = acc[i];
}

// ---------------------------------------------------------------------------
// Launch
// ---------------------------------------------------------------------------
extern "C" void kernel_launch(void* const* d_in, const int* in_sizes, int n_in,
                              void* d_out, int out_size, void* d_ws, size_t ws_size,
                              hipStream_t stream) {
    const float* log_dt     = (const float*)d_in[0];   // (512)
    const float* log_w_real = (const float*)d_in[1];   // (512,32)
    const float* w_imag     = (const float*)d_in[2];   // (512,32)
    const float* Bin        = (const float*)d_in[3];   // (512,32,2)
    const float* Cin        = (const float*)d_in[4];   // (1,512,32,2)
    const float* Pin        = (const float*)d_in[5];   // (1,512,32,2)
    (void)in_sizes; (void)n_in; (void)out_size;

    // workspace layout
    const size_t aElems = (size_t)HH * KP;           // 1,064,960 floats
    const size_t bElems = (size_t)LL * KP;           // 4,259,840 floats
    if (ws_size < (aElems + bElems) * sizeof(float)) return;  // guard
    float* Amat = (float*)d_ws;
    float* Btm  = Amat + aElems;

    // Stage A: transposed basis, one block per output sample l
    s4_fill_basis_t<<<LL, 256, 0, stream>>>(Btm);
    // Stage B: Cauchy + Woodbury, one block per h
    s4_cauchy_woodbury<<<HH, 128, 0, stream>>>(log_dt, log_w_real, w_imag,
                                               Bin, Cin, Pin, Amat);
    // Stage C: WMMA GEMM (4096 waves, 8 waves/block)
    s4_irfft_wmma<<<512, 256, 0, stream>>>(Amat, Btm, (float*)d_out);
}